// MemoryAugmentedNetwork_25718264168585
// MI455X (gfx1250) — compile-verified
//
#include <hip/hip_runtime.h>
#include <stdint.h>

// ---------------------------------------------------------------------------
// MI455X (gfx1250). fp32 everywhere (reference is fp32).
// Matrix ops: V_WMMA_F32_16X16X4_F32. Tile staging: GLOBAL_LOAD_ASYNC_TO_LDS
// (ASYNCcnt-tracked DMA) with double-buffered LDS so copy overlaps WMMA.
// ---------------------------------------------------------------------------

typedef float v2f __attribute__((ext_vector_type(2)));
typedef float v8f __attribute__((ext_vector_type(8)));

#define NEG_BIG (-3.402823466e+38f)

__device__ inline v8f wmma4(v2f a, v2f b, v8f c) {
  // (neg_a, A, neg_b, B, c_mod, C, reuse_a, reuse_b)
  return __builtin_amdgcn_wmma_f32_16x16x4_f32(false, a, false, b, (short)0, c,
                                               false, false);
}

// async global->LDS copy of 16B per lane (GV addressing), ASYNCcnt-tracked
__device__ inline void async_cp16(unsigned lds_off, const void* g) {
  asm volatile("global_load_async_to_lds_b128 %0, %1, off"
               :: "v"(lds_off), "v"((unsigned long long)(uintptr_t)g)
               : "memory");
}
__device__ inline void async_wait0() { asm volatile("s_wait_asynccnt 0x0" ::: "memory"); }
__device__ inline void async_wait6() { asm volatile("s_wait_asynccnt 0x6" ::: "memory"); }

__device__ inline unsigned lds_off(const void* p) {
  return (unsigned)(uintptr_t)p;  // generic LDS ptr: low 32 bits = LDS offset
}

__device__ inline void ins3(float v, int id, float t[3], int ix[3]) {
  if (v > t[0]) { t[2]=t[1]; ix[2]=ix[1]; t[1]=t[0]; ix[1]=ix[0]; t[0]=v; ix[0]=id; }
  else if (v > t[1]) { t[2]=t[1]; ix[2]=ix[1]; t[1]=v; ix[1]=id; }
  else if (v > t[2]) { t[2]=v; ix[2]=id; }
}

// ---------------------------------------------------------------------------
// init: zero barrier counters + h ping-pong buffers (ws is poisoned 0xAA)
// ---------------------------------------------------------------------------
__global__ __launch_bounds__(256) void init_kernel(int* bar, float* hbuf) {
  int i = blockIdx.x * 256 + threadIdx.x;
  if (i < 256) bar[i] = 0;
  if (i < 8192) hbuf[i] = 0.f;
}

// ---------------------------------------------------------------------------
// kscale[n] = importance[n] / max(||keys[n]||, eps)   (wave per key row)
// ---------------------------------------------------------------------------
__global__ __launch_bounds__(256) void kscale_kernel(
    const float* __restrict__ keys, const float* __restrict__ imp,
    float* __restrict__ ksc) {
  int lane = threadIdx.x & 31;
  int n = blockIdx.x * 8 + (threadIdx.x >> 5);
  const float* r = keys + (size_t)n * 256;
  float ss = 0.f;
#pragma unroll
  for (int p = 0; p < 8; ++p) { float v = r[p * 32 + lane]; ss += v * v; }
#pragma unroll
  for (int ofs = 16; ofs > 0; ofs >>= 1) ss += __shfl_xor(ss, ofs);
  if (lane == 0) ksc[n] = imp[n] / fmaxf(sqrtf(ss), 1e-12f);
}

// ---------------------------------------------------------------------------
// in-place row L2-normalize of q [1024][256]   (wave per row)
// ---------------------------------------------------------------------------
__global__ __launch_bounds__(256) void rownorm_kernel(float* __restrict__ q) {
  int lane = threadIdx.x & 31;
  int row = blockIdx.x * 8 + (threadIdx.x >> 5);
  float* r = q + (size_t)row * 256;
  float v[8], ss = 0.f;
#pragma unroll
  for (int p = 0; p < 8; ++p) { v[p] = r[p * 32 + lane]; ss += v[p] * v[p]; }
#pragma unroll
  for (int ofs = 16; ofs > 0; ofs >>= 1) ss += __shfl_xor(ss, ofs);
  float inv = 1.f / fmaxf(sqrtf(ss), 1e-12f);
#pragma unroll
  for (int p = 0; p < 8; ++p) r[p * 32 + lane] = v[p] * inv;
}

// ---------------------------------------------------------------------------
// Generic WMMA-f32 GEMM:  C[M,N] = concat(A1[M,K1],A2[M,K2]) @ W[N,K1+K2]^T
//                                  + bias1[N] + bias2[N]
// Tile: 64x128 per 256-thread WG (8 waves, each 32x32 = 2x2 WMMA tiles), BK=32.
// Double-buffered LDS; tiles staged via global_load_async_to_lds_b128.
// K1,K2 mult of 32; M mult of 64; N mult of 128 (true for all our GEMMs).
// ---------------------------------------------------------------------------
__global__ __launch_bounds__(256) void gemm_wmma_bias(
    const float* __restrict__ A1, int K1,
    const float* __restrict__ A2, int K2,
    const float* __restrict__ W,
    const float* __restrict__ bias1, const float* __restrict__ bias2,
    float* __restrict__ C, int N) {
  __shared__ float Alds[2][64][36];   // pad 36: conflict-free reads, 16B rows
  __shared__ float Blds[2][128][36];
  int tid = threadIdx.x;
  int lane = tid & 31, wave = tid >> 5;
  int wm = (wave >> 2) * 32, wn = (wave & 3) * 32;
  int bm = blockIdx.y * 64, bn = blockIdx.x * 128;
  int ln = lane & 15, hi = lane >> 4, hi2 = hi * 2;
  int Ktot = K1 + K2;
  v8f acc00 = {0,0,0,0,0,0,0,0}, acc01 = {0,0,0,0,0,0,0,0};
  v8f acc10 = {0,0,0,0,0,0,0,0}, acc11 = {0,0,0,0,0,0,0,0};
  int ar = tid >> 3, ac = (tid & 7) * 4;

  auto issue_tile = [&](int kt, int buf) {
    const float* Asrc; int Ks, ko;
    if (kt < K1) { Asrc = A1; Ks = K1; ko = kt; }
    else         { Asrc = A2; Ks = K2; ko = kt - K1; }
    async_cp16(lds_off(&Alds[buf][ar][ac]),      &Asrc[(size_t)(bm + ar)      * Ks + ko + ac]);
    async_cp16(lds_off(&Alds[buf][ar + 32][ac]), &Asrc[(size_t)(bm + ar + 32) * Ks + ko + ac]);
    async_cp16(lds_off(&Blds[buf][ar][ac]),      &W[(size_t)(bn + ar)      * Ktot + kt + ac]);
    async_cp16(lds_off(&Blds[buf][ar + 32][ac]), &W[(size_t)(bn + ar + 32) * Ktot + kt + ac]);
    async_cp16(lds_off(&Blds[buf][ar + 64][ac]), &W[(size_t)(bn + ar + 64) * Ktot + kt + ac]);
    async_cp16(lds_off(&Blds[buf][ar + 96][ac]), &W[(size_t)(bn + ar + 96) * Ktot + kt + ac]);
  };

  int T = Ktot / 32, p = 0;
  issue_tile(0, 0);
  for (int i = 0; i < T; ++i) {
    if (i + 1 < T) { issue_tile((i + 1) * 32, p ^ 1); async_wait6(); }
    else           { async_wait0(); }
    __syncthreads();
#pragma unroll
    for (int kk = 0; kk < 32; kk += 4) {
      v2f a0 = *(const v2f*)&Alds[p][wm + ln][kk + hi2];
      v2f a1 = *(const v2f*)&Alds[p][wm + 16 + ln][kk + hi2];
      v2f b0 = *(const v2f*)&Blds[p][wn + ln][kk + hi2];
      v2f b1 = *(const v2f*)&Blds[p][wn + 16 + ln][kk + hi2];
      acc00 = wmma4(a0, b0, acc00);
      acc01 = wmma4(a0, b1, acc01);
      acc10 = wmma4(a1, b0, acc10);
      acc11 = wmma4(a1, b1, acc11);
    }
    __syncthreads();
    p ^= 1;
  }
  int ng0 = bn + wn + ln, ng1 = ng0 + 16;
  float bv0 = (bias1 ? bias1[ng0] : 0.f) + (bias2 ? bias2[ng0] : 0.f);
  float bv1 = (bias1 ? bias1[ng1] : 0.f) + (bias2 ? bias2[ng1] : 0.f);
#pragma unroll
  for (int r = 0; r < 8; ++r) {
    int mg0 = bm + wm + r + 8 * hi, mg1 = mg0 + 16;
    C[(size_t)mg0 * N + ng0] = acc00[r] + bv0;
    C[(size_t)mg0 * N + ng1] = acc01[r] + bv1;
    C[(size_t)mg1 * N + ng0] = acc10[r] + bv0;
    C[(size_t)mg1 * N + ng1] = acc11[r] + bv1;
  }
}

// ---------------------------------------------------------------------------
// Persistent LSTM: 128 blocks, wave (blockIdx*8+wave) owns hidden unit hi for
// ALL 4 batches -> W_hh rows read once per step. Ping-pong h, per-step global
// barrier via zero-initialized counters (one counter per step -> no reset race)
// ---------------------------------------------------------------------------
__global__ __launch_bounds__(256) void lstm_kernel(
    const float* __restrict__ xproj,  // [B*S][4H], row = b*S+t (biases folded)
    const float* __restrict__ Whh,    // [4H][H], gate order i,f,g,o
    float* __restrict__ hseq,         // [B*S][H]
    float* __restrict__ hbuf,         // [2][B][H] ping-pong
    int* __restrict__ bar) {          // [S] counters (zeroed by init_kernel)
  const int H = 1024, S = 256, NBLK = 128;
  __shared__ float hl[4][1024];
  int tid = threadIdx.x, lane = tid & 31;
  int hi = blockIdx.x * 8 + (tid >> 5);
  const float* w0 = Whh + (size_t)hi * H;
  const float* w1 = Whh + (size_t)(H + hi) * H;
  const float* w2 = Whh + (size_t)(2 * H + hi) * H;
  const float* w3 = Whh + (size_t)(3 * H + hi) * H;
  float c[4] = {0.f, 0.f, 0.f, 0.f};

  for (int t = 0; t < S; ++t) {
    float* hsrc = hbuf + (size_t)(t & 1) * 4 * H;
    for (int i = tid; i < 4 * H; i += 256)
      ((float*)hl)[i] = __hip_atomic_load(hsrc + i, __ATOMIC_RELAXED,
                                          __HIP_MEMORY_SCOPE_AGENT);
    __syncthreads();
    float s0[4] = {0,0,0,0}, s1[4] = {0,0,0,0}, s2[4] = {0,0,0,0}, s3[4] = {0,0,0,0};
    for (int it = 0; it < H; it += 32) {
      int k = it + lane;
      float wa = w0[k], wb = w1[k], wc = w2[k], wd = w3[k];
#pragma unroll
      for (int bb = 0; bb < 4; ++bb) {
        float hv = hl[bb][k];
        s0[bb] = fmaf(wa, hv, s0[bb]);
        s1[bb] = fmaf(wb, hv, s1[bb]);
        s2[bb] = fmaf(wc, hv, s2[bb]);
        s3[bb] = fmaf(wd, hv, s3[bb]);
      }
    }
#pragma unroll
    for (int ofs = 16; ofs > 0; ofs >>= 1) {
#pragma unroll
      for (int bb = 0; bb < 4; ++bb) {
        s0[bb] += __shfl_xor(s0[bb], ofs);
        s1[bb] += __shfl_xor(s1[bb], ofs);
        s2[bb] += __shfl_xor(s2[bb], ofs);
        s3[bb] += __shfl_xor(s3[bb], ofs);
      }
    }
    if (lane == 0) {
      float* hdst = hbuf + (size_t)((t + 1) & 1) * 4 * H;
#pragma unroll
      for (int bb = 0; bb < 4; ++bb) {
        const float* xp = xproj + (size_t)(bb * S + t) * (4 * H);
        float ig = 1.f / (1.f + expf(-(s0[bb] + xp[hi])));
        float fg = 1.f / (1.f + expf(-(s1[bb] + xp[H + hi])));
        float gg = tanhf(s2[bb] + xp[2 * H + hi]);
        float og = 1.f / (1.f + expf(-(s3[bb] + xp[3 * H + hi])));
        c[bb] = fg * c[bb] + ig * gg;
        float hv = og * tanhf(c[bb]);
        hdst[(size_t)bb * H + hi] = hv;
        hseq[(size_t)(bb * S + t) * H + hi] = hv;
      }
    }
    __threadfence();
    __syncthreads();
    if (tid == 0) {
      __hip_atomic_fetch_add(&bar[t], 1, __ATOMIC_ACQ_REL,
                             __HIP_MEMORY_SCOPE_AGENT);
      while (__hip_atomic_load(&bar[t], __ATOMIC_ACQUIRE,
                               __HIP_MEMORY_SCOPE_AGENT) < NBLK)
        __builtin_amdgcn_s_sleep(1);
    }
    __syncthreads();
  }
}

// ---------------------------------------------------------------------------
// Fused sim + per-chunk top-3.  Grid.x = 128 chunks of 512 keys each.
// All 16 row-bands (64 rows) x 4 col-subtiles (128 keys) via the same 64x128
// WMMA-f32 tiling (double-buffered async-LDS staging); tile is dumped to LDS
// (aliased over the staging buffers), scaled by kscale, scanned for top-3.
// Partials written to pvals/pidx [row][chunk][3].
// ---------------------------------------------------------------------------
__global__ __launch_bounds__(256) void sim_topk_kernel(
    const float* __restrict__ qn,      // [1024][256] normalized
    const float* __restrict__ keys,    // [65536][256]
    const float* __restrict__ kscale,  // [65536]
    float* __restrict__ pvals,         // [1024][128][3]
    int* __restrict__ pidx) {
  __shared__ union {
    struct { float A[2][64][36]; float B[2][128][36]; } ab;  // 54 KB
    float ct[64][129];                                       // 33 KB (aliased)
  } u;
  __shared__ float ksl[512];
  __shared__ float tvs[64][4][3];
  __shared__ int   tis[64][4][3];

  int tid = threadIdx.x, lane = tid & 31, wave = tid >> 5;
  int chunk = blockIdx.x;
  int n0 = chunk * 512;
  for (int i = tid; i < 512; i += 256) ksl[i] = kscale[n0 + i];
  int wm = (wave >> 2) * 32, wn = (wave & 3) * 32;
  int ln = lane & 15, hi = lane >> 4, hi2 = hi * 2;
  int ar = tid >> 3, ac = (tid & 7) * 4;
  int srow = tid & 63, squad = tid >> 6;
  __syncthreads();

  for (int band = 0; band < 16; ++band) {
    int bm = band * 64;
    float tv[3] = {NEG_BIG, NEG_BIG, NEG_BIG};
    int ti[3] = {0, 0, 0};
    for (int sub = 0; sub < 4; ++sub) {
      const float* Kbase = keys + (size_t)(n0 + sub * 128) * 256;
      __builtin_prefetch(Kbase + (size_t)ar * 256, 0, 1);  // global_prefetch_b8
      v8f acc00 = {0,0,0,0,0,0,0,0}, acc01 = {0,0,0,0,0,0,0,0};
      v8f acc10 = {0,0,0,0,0,0,0,0}, acc11 = {0,0,0,0,0,0,0,0};

      auto issue_tile = [&](int kt, int buf) {
        async_cp16(lds_off(&u.ab.A[buf][ar][ac]),      &qn[(size_t)(bm + ar)      * 256 + kt + ac]);
        async_cp16(lds_off(&u.ab.A[buf][ar + 32][ac]), &qn[(size_t)(bm + ar + 32) * 256 + kt + ac]);
        async_cp16(lds_off(&u.ab.B[buf][ar][ac]),      &Kbase[(size_t)(ar)      * 256 + kt + ac]);
        async_cp16(lds_off(&u.ab.B[buf][ar + 32][ac]), &Kbase[(size_t)(ar + 32) * 256 + kt + ac]);
        async_cp16(lds_off(&u.ab.B[buf][ar + 64][ac]), &Kbase[(size_t)(ar + 64) * 256 + kt + ac]);
        async_cp16(lds_off(&u.ab.B[buf][ar + 96][ac]), &Kbase[(size_t)(ar + 96) * 256 + kt + ac]);
      };

      int p = 0;
      issue_tile(0, 0);
      for (int i = 0; i < 8; ++i) {           // K = 256 = 8 tiles of 32
        if (i + 1 < 8) { issue_tile((i + 1) * 32, p ^ 1); async_wait6(); }
        else           { async_wait0(); }
        __syncthreads();
#pragma unroll
        for (int kk = 0; kk < 32; kk += 4) {
          v2f a0 = *(const v2f*)&u.ab.A[p][wm + ln][kk + hi2];
          v2f a1 = *(const v2f*)&u.ab.A[p][wm + 16 + ln][kk + hi2];
          v2f b0 = *(const v2f*)&u.ab.B[p][wn + ln][kk + hi2];
          v2f b1 = *(const v2f*)&u.ab.B[p][wn + 16 + ln][kk + hi2];
          acc00 = wmma4(a0, b0, acc00);
          acc01 = wmma4(a0, b1, acc01);
          acc10 = wmma4(a1, b0, acc10);
          acc11 = wmma4(a1, b1, acc11);
        }
        __syncthreads();
        p ^= 1;
      }
      // dump tile over the (now idle) staging buffers and scan for top-3
#pragma unroll
      for (int r = 0; r < 8; ++r) {
        u.ct[wm + r + 8 * hi][wn + ln]           = acc00[r];
        u.ct[wm + r + 8 * hi][wn + 16 + ln]      = acc01[r];
        u.ct[wm + 16 + r + 8 * hi][wn + ln]      = acc10[r];
        u.ct[wm + 16 + r + 8 * hi][wn + 16 + ln] = acc11[r];
      }
      __syncthreads();
      for (int cc = 0; cc < 32; ++cc) {       // thread (srow,squad): 32 cols
        int col = squad * 32 + cc;
        float v = u.ct[srow][col] * ksl[sub * 128 + col];
        ins3(v, n0 + sub * 128 + col, tv, ti);
      }
      __syncthreads();                        // scan done before next staging
    }
    // merge the 4 per-quarter top-3s of each row, write chunk partials
#pragma unroll
    for (int j = 0; j < 3; ++j) { tvs[srow][squad][j] = tv[j]; tis[srow][squad][j] = ti[j]; }
    __syncthreads();
    if (tid < 64) {
      float m[3] = {NEG_BIG, NEG_BIG, NEG_BIG};
      int mi[3] = {0, 0, 0};
#pragma unroll
      for (int qd = 0; qd < 4; ++qd)
#pragma unroll
        for (int j = 0; j < 3; ++j) ins3(tvs[tid][qd][j], tis[tid][qd][j], m, mi);
      int rg = bm + tid;
#pragma unroll
      for (int j = 0; j < 3; ++j) {
        pvals[((size_t)rg * 128 + chunk) * 3 + j] = m[j];
        pidx[((size_t)rg * 128 + chunk) * 3 + j] = mi[j];
      }
    }
    __syncthreads();
  }
}

// ---------------------------------------------------------------------------
// Merge 128 chunk-partials (384 candidates) per row -> final top-3 indices.
// Wave per row.
// ---------------------------------------------------------------------------
__global__ __launch_bounds__(256) void topk_merge_kernel(
    const float* __restrict__ pvals, const int* __restrict__ pidx,
    int* __restrict__ topidx) {
  int lane = threadIdx.x & 31;
  int row = blockIdx.x * 8 + (threadIdx.x >> 5);
  float t[3] = {NEG_BIG, NEG_BIG, NEG_BIG};
  int ix[3] = {0, 0, 0};
  const float* pv = pvals + (size_t)row * 384;
  const int* pi = pidx + (size_t)row * 384;
#pragma unroll
  for (int j = 0; j < 12; ++j) {
    int p = lane * 12 + j;
    ins3(pv[p], pi[p], t, ix);
  }
#pragma unroll
  for (int ofs = 16; ofs > 0; ofs >>= 1) {
    float o[3]; int oi[3];
#pragma unroll
    for (int j = 0; j < 3; ++j) { o[j] = __shfl_xor(t[j], ofs); oi[j] = __shfl_xor(ix[j], ofs); }
    float a[3]; int ai[3]; float bv[3]; int bi[3];
    if ((lane & ofs) == 0) {
#pragma unroll
      for (int j = 0; j < 3; ++j) { a[j] = t[j]; ai[j] = ix[j]; bv[j] = o[j]; bi[j] = oi[j]; }
    } else {
#pragma unroll
      for (int j = 0; j < 3; ++j) { a[j] = o[j]; ai[j] = oi[j]; bv[j] = t[j]; bi[j] = ix[j]; }
    }
#pragma unroll
    for (int j = 0; j < 3; ++j) { t[j] = a[j]; ix[j] = ai[j]; }
#pragma unroll
    for (int j = 0; j < 3; ++j) ins3(bv[j], bi[j], t, ix);
  }
  if (lane == 0) {
    topidx[row * 3 + 0] = ix[0];
    topidx[row * 3 + 1] = ix[1];
    topidx[row * 3 + 2] = ix[2];
  }
}

// ---------------------------------------------------------------------------
// Attention over 3 retrieved value rows + weighted sum -> mem [1024][256]
// Wave per row.
// ---------------------------------------------------------------------------
__global__ __launch_bounds__(256) void att_mem_kernel(
    const float* __restrict__ values, const float* __restrict__ Wa,
    const float* __restrict__ ba, const int* __restrict__ topidx,
    float* __restrict__ mem) {
  int lane = threadIdx.x & 31;
  int row = blockIdx.x * 8 + (threadIdx.x >> 5);
  float wa[8];
#pragma unroll
  for (int p = 0; p < 8; ++p) wa[p] = Wa[p * 32 + lane];
  float vr[3][8], lg[3];
#pragma unroll
  for (int j = 0; j < 3; ++j) {
    int idx = topidx[row * 3 + j];
    const float* v = values + (size_t)idx * 256;
    float s = 0.f;
#pragma unroll
    for (int p = 0; p < 8; ++p) { vr[j][p] = v[p * 32 + lane]; s = fmaf(vr[j][p], wa[p], s); }
#pragma unroll
    for (int ofs = 16; ofs > 0; ofs >>= 1) s += __shfl_xor(s, ofs);
    lg[j] = s + ba[0];
  }
  float mx = fmaxf(lg[0], fmaxf(lg[1], lg[2]));
  float e0 = expf(lg[0] - mx), e1 = expf(lg[1] - mx), e2 = expf(lg[2] - mx);
  float inv = 1.f / (e0 + e1 + e2);
  float a0 = e0 * inv, a1 = e1 * inv, a2 = e2 * inv;
#pragma unroll
  for (int p = 0; p < 8; ++p)
    mem[(size_t)row * 256 + p * 32 + lane] =
        a0 * vr[0][p] + a1 * vr[1][p] + a2 * vr[2][p];
}

// ---------------------------------------------------------------------------
extern "C" void kernel_launch(void* const* d_in, const int* in_sizes, int n_in,
                              void* d_out, int out_size, void* d_ws,
                              size_t ws_size, hipStream_t stream) {
  (void)in_sizes; (void)n_in; (void)out_size; (void)ws_size;
  const float* x          = (const float*)d_in[0];
  const float* keys       = (const float*)d_in[1];
  const float* values     = (const float*)d_in[2];
  const float* importance = (const float*)d_in[3];
  const float* W_ih       = (const float*)d_in[4];
  const float* W_hh       = (const float*)d_in[5];
  const float* b_ih       = (const float*)d_in[6];
  const float* b_hh       = (const float*)d_in[7];
  const float* Wq         = (const float*)d_in[8];
  const float* bq         = (const float*)d_in[9];
  const float* Wa         = (const float*)d_in[10];
  const float* ba         = (const float*)d_in[11];
  const float* Wc         = (const float*)d_in[12];
  const float* bc         = (const float*)d_in[13];
  const float* Wo         = (const float*)d_in[14];
  const float* bo         = (const float*)d_in[15];
  float* out = (float*)d_out;

  // workspace layout (~27.6 MB)
  float* ws     = (float*)d_ws;
  float* xproj  = ws;                   // 1024*4096
  float* hseq   = xproj + 4194304;      // 1024*1024
  float* hbuf   = hseq + 1048576;       // 2*4*1024
  float* q      = hbuf + 8192;          // 1024*256
  float* kscale = q + 262144;           // 65536
  float* pvals  = kscale + 65536;       // 1024*128*3
  float* mem    = pvals + 393216;       // 1024*256
  float* mem2   = mem + 262144;         // 1024*256
  int* pidx   = (int*)(mem2 + 262144);  // 1024*128*3
  int* topidx = pidx + 393216;          // 1024*3
  int* bar    = topidx + 3072;          // 256

  init_kernel<<<32, 256, 0, stream>>>(bar, hbuf);
  kscale_kernel<<<8192, 256, 0, stream>>>(keys, importance, kscale);
  // x_proj = x2d @ W_ih^T + (b_ih + b_hh)    [1024 x 4096], K=512
  gemm_wmma_bias<<<dim3(32, 16), 256, 0, stream>>>(
      x, 512, nullptr, 0, W_ih, b_ih, b_hh, xproj, 4096);
  // sequential LSTM recurrence (persistent, per-step global barrier)
  lstm_kernel<<<128, 256, 0, stream>>>(xproj, W_hh, hseq, hbuf, bar);
  // q = hseq @ Wq^T + bq                     [1024 x 256], K=1024
  gemm_wmma_bias<<<dim3(2, 16), 256, 0, stream>>>(
      hseq, 1024, nullptr, 0, Wq, bq, nullptr, q, 256);
  rownorm_kernel<<<128, 256, 0, stream>>>(q);
  // fused cosine-sim (34 GF, WMMA f32) + per-chunk top-3
  sim_topk_kernel<<<128, 256, 0, stream>>>(q, keys, kscale, pvals, pidx);
  topk_merge_kernel<<<128, 256, 0, stream>>>(pvals, pidx, topidx);
  att_mem_kernel<<<128, 256, 0, stream>>>(values, Wa, ba, topidx, mem);
  // mem2 = mem @ Wc^T + bc                   [1024 x 256], K=256
  gemm_wmma_bias<<<dim3(2, 16), 256, 0, stream>>>(
      mem, 256, nullptr, 0, Wc, bc, nullptr, mem2, 256);
  // out = concat(hseq, mem2) @ Wo^T + bo     [1024 x 512], K=1024+256
  gemm_wmma_bias<<<dim3(4, 16), 256, 0, stream>>>(
      hseq, 1024, mem2, 256, Wo, bo, nullptr, out, 512);
}